// EncodingLayerFilter_45294725103998
// MI455X (gfx1250) — compile-verified
//
#include <hip/hip_runtime.h>
#include <hip/hip_bf16.h>
#include <math.h>

typedef __attribute__((ext_vector_type(2))) float v2f;
typedef __attribute__((ext_vector_type(8))) float v8f;

#define NF     10000
#define NTILE  (NF / 16)          /* 625 */
#define PATCH  16
#define ENC    128
#define BDIM   8
#define HDIM   64
#define WDIM   32
#define NPOS   (BDIM*HDIM*WDIM)   /* 16384 */
#define EPSV   1e-8f

// ---------------------------------------------------------------------------
// Kernel 1: f_sum[n] = sum_p filters[n][p]
// ---------------------------------------------------------------------------
__global__ __launch_bounds__(256) void k_fsum(const float* __restrict__ filters,
                                              float* __restrict__ f_sum) {
  int n = blockIdx.x * 256 + threadIdx.x;
  if (n >= NF) return;
  const float* f = filters + n * PATCH;
  float s = 0.f;
  #pragma unroll
  for (int p = 0; p < PATCH; ++p) s += f[p];
  f_sum[n] = s;
}

// ---------------------------------------------------------------------------
// Kernel 2: min-max normalize over H (axis=1), then sum over patch dim.
// One 64-thread block per (b,w).
// ---------------------------------------------------------------------------
__global__ __launch_bounds__(64) void k_xsum(const float* __restrict__ x,
                                             float* __restrict__ x_sum) {
  __shared__ float mn[PATCH], mx[PATCH];
  int blk = blockIdx.x;          // 0..255
  int b = blk >> 5;
  int w = blk & 31;
  int tid = threadIdx.x;         // 0..63

  if (tid < PATCH) {
    int p = tid;
    float lo = INFINITY, hi = -INFINITY;
    for (int h = 0; h < HDIM; ++h) {
      float v = x[(((b * HDIM + h) * WDIM + w) * PATCH) + p];
      lo = fminf(lo, v);
      hi = fmaxf(hi, v);
    }
    mn[p] = lo; mx[p] = hi;
  }
  __syncthreads();

  int h = tid;
  const float* row = x + ((b * HDIM + h) * WDIM + w) * PATCH;
  float s = 0.f;
  #pragma unroll
  for (int p = 0; p < PATCH; ++p)
    s += (row[p] - mn[p]) / (mx[p] - mn[p] + EPSV);
  x_sum[(b * HDIM + h) * WDIM + w] = s;
}

// ---------------------------------------------------------------------------
// Kernel 3: argmin_n |f_sum[n] - x_sum[pos]| via V_WMMA_F32_16X16X4_F32.
//   A row n : [f_sum[n], 1, *, *]   (K=2,3 operands multiplied by zero B rows)
//   B col p : [1, -x_sum[pos], 0, 0]^T
//   D[n,pos] = f - x   (bit-exact fp32 subtraction)
// Software-pipelined LDS fetch (1 tile ahead); 8 independent argmin
// accumulators (one per C register) that record only the iteration number;
// global index reconstructed after the loop; lexicographic (d, n) merges
// preserve first-index tie-break.
// ---------------------------------------------------------------------------
__global__ __launch_bounds__(256) void k_argmin(const float* __restrict__ f_sum,
                                                const float* __restrict__ x_sum,
                                                int* __restrict__ idx_out) {
  __shared__ float sf[NF + 16];               // +16: prefetch overrun pad
  int tid = threadIdx.x;
  for (int j = tid; j < NF + 16; j += 256) sf[j] = (j < NF) ? f_sum[j] : 0.f;
  __syncthreads();

  int wave = tid >> 5;
  int lane = tid & 31;
  int np   = lane & 15;
  bool lo  = lane < 16;
  int pos  = blockIdx.x * 128 + wave * 16 + np;

  float xs = x_sum[pos];

  v2f B;
  B.x = lo ? 1.0f : 0.0f;   // K=0 row (lo lanes) / K=2 row (hi lanes, zero)
  B.y = lo ? -xs  : 0.0f;   // K=1 row            / K=3 row (zero)
  int mbase = lo ? 0 : 8;   // C layout: VGPR r -> M=r (lo) / M=r+8 (hi)

  float run_d[8];
  int   run_it[8];
  #pragma unroll
  for (int r = 0; r < 8; ++r) { run_d[r] = INFINITY; run_it[r] = 0; }

  float fa = sf[np];                          // tile 0, prefetched
  for (int i = 0; i < NTILE; ++i) {
    v2f A;
    A.x = fa;                                 // all lanes: K>=2 rows of B are 0
    A.y = 1.0f;
    float fa_n = sf[(i + 1) * 16 + np];       // prefetch next tile (pad-safe)
    v8f C = {};
    C = __builtin_amdgcn_wmma_f32_16x16x4_f32(
        false, A, false, B, (short)0, C, false, false);
    #pragma unroll
    for (int r = 0; r < 8; ++r) {             // 8 independent chains
      float d = __builtin_fabsf(C[r]);
      bool lt = d < run_d[r];                 // strict: keeps earliest i
      run_d[r]  = lt ? d : run_d[r];
      run_it[r] = lt ? i : run_it[r];
    }
    fa = fa_n;
  }

  // Per-lane merge of the 8 accumulators: lexicographic (d, n) min.
  float bd = INFINITY;
  int   bn = 0x7fffffff;
  #pragma unroll
  for (int r = 0; r < 8; ++r) {
    int n = run_it[r] * 16 + mbase + r;
    bool t = (run_d[r] < bd) || (run_d[r] == bd && n < bn);
    bd = t ? run_d[r] : bd;
    bn = t ? n : bn;
  }

  // Merge the two half-wave candidates for each position.
  float od = __shfl_xor(bd, 16, 32);
  int   on = __shfl_xor(bn, 16, 32);
  if (od < bd || (od == bd && on < bn)) { bd = od; bn = on; }

  if (lo) idx_out[pos] = bn;
}

// ---------------------------------------------------------------------------
// Kernel 4: out[pos][:] = emb[idx[pos]][:]  (float4 coalesced, 32 f4 / pos)
// ---------------------------------------------------------------------------
__global__ __launch_bounds__(256) void k_gather(const float* __restrict__ emb,
                                                const int* __restrict__ idx,
                                                float* __restrict__ out) {
  int t   = blockIdx.x * 256 + threadIdx.x;  // 0 .. NPOS*ENC/4 - 1
  int pos = t >> 5;                          // ENC/4 = 32 float4 per position
  int r   = t & 31;
  const float4* e4 = (const float4*)emb;
  float4*       o4 = (float4*)out;
  o4[t] = e4[(size_t)idx[pos] * 32 + r];
}

// ---------------------------------------------------------------------------
extern "C" void kernel_launch(void* const* d_in, const int* in_sizes, int n_in,
                              void* d_out, int out_size, void* d_ws, size_t ws_size,
                              hipStream_t stream) {
  const float* x       = (const float*)d_in[0];  // (8,64,32,16)
  const float* filters = (const float*)d_in[1];  // (10000,16)
  const float* emb     = (const float*)d_in[2];  // (10000,128)
  float* out = (float*)d_out;                    // (8,64,32,128)

  float* ws    = (float*)d_ws;
  float* x_sum = ws;                             // 16384 floats
  float* f_sum = ws + NPOS;                      // 10000 floats (pad to 10016)
  int*   idx   = (int*)(ws + NPOS + 10016);      // 16384 ints

  k_fsum  <<<(NF + 255) / 256, 256, 0, stream>>>(filters, f_sum);
  k_xsum  <<<BDIM * WDIM,       64, 0, stream>>>(x, x_sum);
  k_argmin<<<NPOS / 128,       256, 0, stream>>>(f_sum, x_sum, idx);
  k_gather<<<(NPOS * ENC / 4) / 256, 256, 0, stream>>>(emb, idx, out);
}